// PreProcessing_11484742549760
// MI455X (gfx1250) — compile-verified
//
#include <hip/hip_runtime.h>
#include <cstdint>

#define AS1 __attribute__((address_space(1)))
#define AS3 __attribute__((address_space(3)))

// ---------------- CDNA5 async global->LDS helpers ----------------
__device__ __forceinline__ void async_load_f32_to_lds(const float* gptr, uint32_t lds_byte_off) {
#if __has_builtin(__builtin_amdgcn_global_load_async_to_lds_b32)
    __builtin_amdgcn_global_load_async_to_lds_b32(
        (AS1 int*)(uintptr_t)gptr,
        (AS3 int*)(uintptr_t)lds_byte_off,
        /*offset=*/0, /*cpol=*/0);
#else
    asm volatile("global_load_async_to_lds_b32 %0, %1, off"
                 :: "v"(lds_byte_off), "v"((uint64_t)(uintptr_t)gptr)
                 : "memory");
#endif
}

__device__ __forceinline__ void wait_asynccnt0() {
#if __has_builtin(__builtin_amdgcn_s_wait_asynccnt)
    __builtin_amdgcn_s_wait_asynccnt(0);
#else
    asm volatile("s_wait_asynccnt 0x0" ::: "memory");
#endif
}

// ---------------- problem constants ----------------
constexpr int   W_IMG = 512;
constexpr int   H_IMG = 512;
constexpr int   TW    = 32;          // tile width  (outputs)
constexpr int   THT   = 32;          // tile height (outputs)
constexpr int   BY    = 8;           // blockDim.y  -> 32x8 = 256 threads = 8 waves
constexpr int   RPT   = THT / BY;    // 4 rows per thread
constexpr int   HWD   = TW + 4;      // 36 halo width
constexpr int   HHT   = THT + 4;     // 36 halo height
constexpr int   HELEM = HWD * HHT;   // 1296 halo elements

// color_coeff = -0.5/0.05^2 = -200 ; exp(c*d2) = exp2(C2*d2), C2 = -200*log2(e)
constexpr float C2  = -288.53900817779268f;
// spatial weights exp(-0.5*d2/2500)
constexpr float SW1 = 0.99980001999866674f;  // d2=1
constexpr float SW2 = 0.99960007998933380f;  // d2=2
constexpr float SW4 = 0.99920031991466671f;  // d2=4

__device__ __forceinline__ int reflect_idx(int g, int n) {
    // jnp.pad 'reflect': -1 -> 1, -2 -> 2 ; n -> n-2, n+1 -> n-3  (radius<=2, n>=4)
    g = (g < 0) ? -g : g;
    g = (g >= n) ? (2 * n - 2 - g) : g;
    return g;
}

__global__ __launch_bounds__(256)
void bilateral_enhance_kernel(const float* __restrict__ x, float* __restrict__ out) {
    __shared__ float tile[HELEM];

    const int tid = threadIdx.y * 32 + threadIdx.x;
    const int x0  = blockIdx.x * TW;
    const int y0  = blockIdx.y * THT;
    const float* img = x + (size_t)blockIdx.z * (H_IMG * W_IMG);

    const uint32_t lds_base = (uint32_t)(uintptr_t)&tile[0]; // LDS byte offset

    // ---- stage halo tile (raw x) into LDS via async DMA loads (ASYNCcnt) ----
    for (int i = tid; i < HELEM; i += 256) {
        const int ly = i / HWD;
        const int lx = i - ly * HWD;
        const int gx = reflect_idx(x0 - 2 + lx, W_IMG);
        const int gy = reflect_idx(y0 - 2 + ly, H_IMG);
        async_load_f32_to_lds(img + gy * W_IMG + gx, lds_base + (uint32_t)i * 4u);
    }
    wait_asynccnt0();
    __syncthreads();

    // ---- tone-map in place: v=clip((x+1)/2); blended = 0.6*v + 0.4*v^1.5 = v*(0.6+0.4*sqrt(v))
#pragma unroll
    for (int k = 0; k < (HELEM + 255) / 256; ++k) {
        const int i = tid + k * 256;
        if (i < HELEM) {
            float v = tile[i];
            v = (v + 1.0f) * 0.5f;
            v = fminf(fmaxf(v, 1e-6f), 1.0f);
            tile[i] = v * fmaf(0.4f, __builtin_amdgcn_sqrtf(v), 0.6f);
        }
    }
    __syncthreads();

    // ---- 13-tap bilateral from LDS, 4 output rows per thread (fully unrolled
    //      -> ~52 independent v_exp_f32 chains per thread to hide TRANS latency) ----
    const int lx = threadIdx.x + 2;
#pragma unroll
    for (int r = 0; r < RPT; ++r) {
        const int py  = (int)threadIdx.y + r * BY;  // tile-local output row
        const int lyc = py + 2;
        const float c = tile[lyc * HWD + lx];
        float num = 0.0f, den = 0.0f;

#define TAP(DY, DX, SW)                                                     \
        { float s = tile[(lyc + (DY)) * HWD + lx + (DX)];                   \
          float d = s - c;                                                  \
          float w = (SW) * __builtin_amdgcn_exp2f(C2 * d * d);              \
          num = fmaf(w, s, num); den += w; }

        TAP(-2,  0, SW4)
        TAP(-1, -1, SW2) TAP(-1, 0, SW1) TAP(-1, 1, SW2)
        TAP( 0, -2, SW4) TAP( 0,-1, SW1) TAP( 0, 0, 1.0f) TAP(0, 1, SW1) TAP(0, 2, SW4)
        TAP( 1, -1, SW2) TAP( 1, 0, SW1) TAP( 1, 1, SW2)
        TAP( 2,  0, SW4)
#undef TAP

        out[((size_t)blockIdx.z * H_IMG + (y0 + py)) * W_IMG + (x0 + threadIdx.x)] =
            num * __builtin_amdgcn_rcpf(den);
    }
}

extern "C" void kernel_launch(void* const* d_in, const int* in_sizes, int n_in,
                              void* d_out, int out_size, void* d_ws, size_t ws_size,
                              hipStream_t stream) {
    (void)n_in; (void)d_ws; (void)ws_size; (void)out_size;
    const float* x = (const float*)d_in[0];
    float* out     = (float*)d_out;
    const int B    = in_sizes[0] / (H_IMG * W_IMG);   // 32
    dim3 grid(W_IMG / TW, H_IMG / THT, B);
    dim3 block(32, BY, 1);
    hipLaunchKernelGGL(bilateral_enhance_kernel, grid, block, 0, stream, x, out);
}